// EncoderLayer_33457795236558
// MI455X (gfx1250) — compile-verified
//
#include <hip/hip_runtime.h>
#include <math.h>

#define DD 16
#define LL 256
#define HH 128
#define BB 8
#define LDK 17
#define LDH 132   // 132 floats = 528B: rows 16B-aligned for b128 LDS loads, conflict-free banks
#define LDL 257
#define EPSF 1e-6f

typedef float v2f __attribute__((ext_vector_type(2)));
typedef float v8f __attribute__((ext_vector_type(8)));

__device__ __forceinline__ v8f wmma4(v2f a, v2f b, v8f c) {
  // D = A(16x4) * B(4x16) + C, all fp32 (exact math vs fp32 reference)
  return __builtin_amdgcn_wmma_f32_16x16x4_f32(false, a, false, b, (short)0, c,
                                               false, false);
}

// 16x16 tile GEMM: D = A[16xK] * B[Kx16] + Cinit (K multiple of 4).
// A-frag (16x4, 2 VGPR): lane m=lane&15 holds rows, lane bit4 selects k-pair.
// B-frag mirrors A with n=lane&15 as the column.
__device__ __forceinline__ v8f gemm_tile(const float* A, int lda,
                                         const float* Bm, int ldb,
                                         int K, v8f c) {
  const int lane = threadIdx.x & 31;
  const int mn = lane & 15;
  const int kk = (lane >> 4) << 1;
  for (int k = 0; k < K; k += 4) {
    v2f a, b;
    a.x = A[mn * lda + k + kk];
    a.y = A[mn * lda + k + kk + 1];
    b.x = Bm[(k + kk) * ldb + mn];
    b.y = Bm[(k + kk + 1) * ldb + mn];
    c = wmma4(a, b, c);
  }
  return c;
}

// C/D layout: VGPR r -> row (r + 8*(lane>=16)), col = lane&15
__device__ __forceinline__ void store_tile(float* dst, int ldd, v8f c) {
  const int lane = threadIdx.x & 31;
  const int n = lane & 15;
  const int rbase = (lane >> 4) << 3;
#pragma unroll
  for (int r = 0; r < 8; ++r) dst[(r + rbase) * ldd + n] = c[r];
}

__device__ __forceinline__ v8f splat8(float v) {
  v8f c;
#pragma unroll
  for (int r = 0; r < 8; ++r) c[r] = v;
  return c;
}

__global__ __launch_bounds__(256) void enc_layer_kernel(
    const float* __restrict__ x, const float* __restrict__ mask,
    const float* __restrict__ wq, const float* __restrict__ bq,
    const float* __restrict__ wk, const float* __restrict__ bk,
    const float* __restrict__ wv, const float* __restrict__ bv,
    const float* __restrict__ wo, const float* __restrict__ bo,
    const float* __restrict__ w1, const float* __restrict__ b1,
    const float* __restrict__ w2, const float* __restrict__ b2,
    const float* __restrict__ f1, const float* __restrict__ f1b,
    const float* __restrict__ f2, const float* __restrict__ f2b,
    const float* __restrict__ g1, const float* __restrict__ be1,
    const float* __restrict__ g2, const float* __restrict__ be2,
    float* __restrict__ out) {
  __shared__ __align__(16) float ks[LL * LDK];   // k for all 256 rows
  __shared__ __align__(16) float vs[LL * LDK];   // v for all 256 rows
  __shared__ __align__(16) float qs[DD * LDK];   // q for this i-tile
  __shared__ __align__(16) float qa[DD * LDH];   // q @ w1q
  __shared__ __align__(16) float qb[DD * LDH];   // q @ w1k
  __shared__ __align__(16) float kbc[DD * LDH];  // chunk: k @ w1k + b1
  __shared__ __align__(16) float kac[DD * LDH];  // chunk: k @ w1q + b1
  __shared__ __align__(16) float lg[DD * LDL];   // logits / attn (16 x 256)
  __shared__ __align__(16) float w2s[HH];
  __shared__ __align__(16) float part[8 * DD * DD];  // per-wave partial tiles
  __shared__ __align__(16) float ctx[DD * LDK];
  __shared__ __align__(16) float aout[DD * LDK];
  __shared__ __align__(16) float o1[DD * LDK];
  __shared__ __align__(16) float hid[DD * LDH];

  const int tid = threadIdx.x;
  const int wave = tid >> 5;
  const int lane = tid & 31;
  const int nfr = lane & 15;         // fragment column index
  const int bidx = blockIdx.x >> 4;  // batch
  const int i0 = (blockIdx.x & 15) * DD;

  const float* xb = x + bidx * LL * DD;

  if (tid < HH / 4) ((float4*)w2s)[tid] = ((const float4*)w2)[tid];

  // ---- Phase 1: k,v for all rows; q for i-tile (33 WMMA tile-jobs / 8 waves)
  for (int job = wave; job < 33; job += 8) {
    const float* W;
    const float* bias;
    int r0;
    float* dst;
    if (job < 16) {
      W = wk; bias = bk; r0 = job * 16; dst = &ks[r0 * LDK];
    } else if (job < 32) {
      W = wv; bias = bv; r0 = (job - 16) * 16; dst = &vs[r0 * LDK];
    } else {
      W = wq; bias = bq; r0 = i0; dst = qs;
    }
    v8f c = splat8(bias[nfr]);
    c = gemm_tile(xb + r0 * DD, DD, W, DD, DD, c);
    store_tile(dst, LDK, c);
  }
  __syncthreads();

  // ---- Phase 2: qa = q@w1q, qb = q@w1k (16 h-tile jobs)
  for (int job = wave; job < 16; job += 8) {
    const int ht = job & 7;
    const float* Bm = w1 + (job < 8 ? 0 : DD * HH) + ht * 16;
    float* dst = (job < 8 ? qa : qb) + ht * 16;
    v8f c = splat8(0.0f);
    c = gemm_tile(qs, LDK, Bm, HH, DD, c);
    store_tile(dst, LDH, c);
  }
  __syncthreads();

  const float b2v = b2[0];
  const int ii = tid >> 4;  // row in i-tile (0..15)
  const int jj = tid & 15;  // col within chunk / tile (0..15)
  const float* mrow = mask + (size_t)(bidx * LL + i0 + ii) * LL + jj;

  // ---- Phase 3: pairwise attention-MLP logits, 16 j per chunk
  for (int jc = 0; jc < 16; ++jc) {
    const int j0 = jc * 16;
    // prefetch next chunk's mask element while we compute this chunk
    if (jc + 1 < 16) __builtin_prefetch(mrow + j0 + 16, 0, 0);

    // kbc = k@w1k + b1 ; kac = k@w1q + b1  (b1 folded into C)
    for (int job = wave; job < 16; job += 8) {
      const int ht = job & 7;
      const float* Bm = w1 + (job < 8 ? DD * HH : 0) + ht * 16;
      float* dst = (job < 8 ? kbc : kac) + ht * 16;
      v8f c = splat8(b1[ht * 16 + nfr]);
      c = gemm_tile(&ks[j0 * LDK], LDK, Bm, HH, DD, c);
      store_tile(dst, LDH, c);
    }
    __syncthreads();

    // one (i,j) pair per thread: relu-MLP dot with w2, b128 LDS reads
    {
      const float4* pqa = (const float4*)&qa[ii * LDH];
      const float4* pqb = (const float4*)&qb[ii * LDH];
      const float4* pkb = (const float4*)&kbc[jj * LDH];
      const float4* pka = (const float4*)&kac[jj * LDH];
      const float4* pw2 = (const float4*)w2s;
      float s = 0.0f;
#pragma unroll 4
      for (int h4 = 0; h4 < HH / 4; ++h4) {
        const float4 A = pqa[h4];
        const float4 Bq = pqb[h4];
        const float4 Kb = pkb[h4];
        const float4 Ka = pka[h4];
        const float4 W = pw2[h4];
        s += fmaxf(A.x + Kb.x, 0.0f) * W.x;
        s += fmaxf(A.y + Kb.y, 0.0f) * W.y;
        s += fmaxf(A.z + Kb.z, 0.0f) * W.z;
        s += fmaxf(A.w + Kb.w, 0.0f) * W.w;
        s += fmaxf(Bq.x + Ka.x, 0.0f) * W.x;
        s += fmaxf(Bq.y + Ka.y, 0.0f) * W.y;
        s += fmaxf(Bq.z + Ka.z, 0.0f) * W.z;
        s += fmaxf(Bq.w + Ka.w, 0.0f) * W.w;
      }
      s += 2.0f * b2v;
      s += mrow[j0] * -1e9f;
      lg[ii * LDL + j0 + jj] = s;
    }
    __syncthreads();
  }

  // ---- Phase 4: softmax over j (2 rows per wave, lane-parallel, wave32 shfl)
  for (int r = wave; r < 16; r += 8) {
    float m = -3.0e38f;
    for (int c = lane; c < LL; c += 32) m = fmaxf(m, lg[r * LDL + c]);
#pragma unroll
    for (int off = 16; off >= 1; off >>= 1)
      m = fmaxf(m, __shfl_xor(m, off, 32));
    float ssum = 0.0f;
    for (int c = lane; c < LL; c += 32) {
      const float e = __expf(lg[r * LDL + c] - m);
      lg[r * LDL + c] = e;
      ssum += e;
    }
#pragma unroll
    for (int off = 16; off >= 1; off >>= 1) ssum += __shfl_xor(ssum, off, 32);
    const float inv = 1.0f / ssum;
    for (int c = lane; c < LL; c += 32) lg[r * LDL + c] *= inv;
  }
  __syncthreads();

  // ---- Phase 5: ctx = attn @ v, K=256 split over 8 waves (8 WMMAs each)
  {
    v8f c = splat8(0.0f);
    c = gemm_tile(&lg[wave * 32], LDL, &vs[wave * 32 * LDK], LDK, 32, c);
    store_tile(&part[wave * DD * DD], DD, c);
  }
  __syncthreads();
  {
    float acc = 0.0f;
#pragma unroll
    for (int w = 0; w < 8; ++w) acc += part[w * DD * DD + ii * DD + jj];
    ctx[ii * LDK + jj] = acc;
  }
  __syncthreads();

  // attn_out = ctx @ wo + bo
  if (wave == 0) {
    v8f c = splat8(bo[nfr]);
    c = gemm_tile(ctx, LDK, wo, DD, DD, c);
    store_tile(aout, LDK, c);
  }
  __syncthreads();

  // ---- LayerNorm 1 (RES=1): out1 = LN(x + attn_out); 16-lane shfl reduce
  {
    const float y = xb[(i0 + ii) * DD + jj] + aout[ii * LDK + jj];
    float s = y;
#pragma unroll
    for (int off = 8; off >= 1; off >>= 1) s += __shfl_xor(s, off, 32);
    const float mean = s * (1.0f / 16.0f);
    const float d = y - mean;
    float v = d * d;
#pragma unroll
    for (int off = 8; off >= 1; off >>= 1) v += __shfl_xor(v, off, 32);
    o1[ii * LDK + jj] = d * rsqrtf(v * (1.0f / 16.0f) + EPSF) * g1[jj] + be1[jj];
  }
  __syncthreads();

  // ---- Phase 6: hid = relu(o1 @ f1 + f1b), one h-tile per wave
  {
    const int ht = wave;
    v8f c = splat8(f1b[ht * 16 + nfr]);
    c = gemm_tile(o1, LDK, f1 + ht * 16, HH, DD, c);
    const int rbase = (lane >> 4) << 3;
#pragma unroll
    for (int r = 0; r < 8; ++r)
      hid[(r + rbase) * LDH + ht * 16 + nfr] = fmaxf(c[r], 0.0f);
  }
  __syncthreads();

  // ffn = hid @ f2 + f2b, K=128 split over 8 waves
  {
    v8f c = splat8(0.0f);
    c = gemm_tile(hid + wave * 16, LDH, f2 + wave * 16 * DD, DD, 16, c);
    store_tile(&part[wave * DD * DD], DD, c);
  }
  __syncthreads();

  // reduce partials + LayerNorm 2 + store
  {
    float ffn = f2b[jj];
#pragma unroll
    for (int w = 0; w < 8; ++w) ffn += part[w * DD * DD + ii * DD + jj];
    const float y = o1[ii * LDK + jj] + ffn;  // RES = 1
    float s = y;
#pragma unroll
    for (int off = 8; off >= 1; off >>= 1) s += __shfl_xor(s, off, 32);
    const float mean = s * (1.0f / 16.0f);
    const float d = y - mean;
    float v = d * d;
#pragma unroll
    for (int off = 8; off >= 1; off >>= 1) v += __shfl_xor(v, off, 32);
    out[(bidx * LL + i0 + ii) * DD + jj] =
        d * rsqrtf(v * (1.0f / 16.0f) + EPSF) * g2[jj] + be2[jj];
  }
}

extern "C" void kernel_launch(void* const* d_in, const int* in_sizes, int n_in,
                              void* d_out, int out_size, void* d_ws,
                              size_t ws_size, hipStream_t stream) {
  (void)in_sizes; (void)n_in; (void)d_ws; (void)ws_size; (void)out_size;
  const float* x    = (const float*)d_in[0];
  const float* mask = (const float*)d_in[1];
  const float* wq   = (const float*)d_in[2];
  const float* bq   = (const float*)d_in[3];
  const float* wk   = (const float*)d_in[4];
  const float* bk   = (const float*)d_in[5];
  const float* wv   = (const float*)d_in[6];
  const float* bv   = (const float*)d_in[7];
  const float* wo   = (const float*)d_in[8];
  const float* bo   = (const float*)d_in[9];
  const float* w1   = (const float*)d_in[10];
  const float* b1   = (const float*)d_in[11];
  const float* w2   = (const float*)d_in[12];
  const float* b2   = (const float*)d_in[13];
  const float* f1   = (const float*)d_in[14];
  const float* f1b  = (const float*)d_in[15];
  const float* f2   = (const float*)d_in[16];
  const float* f2b  = (const float*)d_in[17];
  const float* g1   = (const float*)d_in[18];
  const float* be1  = (const float*)d_in[19];
  const float* g2   = (const float*)d_in[20];
  const float* be2  = (const float*)d_in[21];
  float* out = (float*)d_out;

  dim3 grid(BB * (LL / DD));  // 8 batches x 16 i-tiles = 128 blocks
  dim3 block(256);            // 8 wave32s
  enc_layer_kernel<<<grid, block, 0, stream>>>(
      x, mask, wq, bq, wk, bk, wv, bv, wo, bo, w1, b1, w2, b2, f1, f1b, f2,
      f2b, g1, be1, g2, be2, out);
}